// HiPPO_FouT_77266461655042
// MI455X (gfx1250) — compile-verified
//
#include <hip/hip_runtime.h>

// HiPPO-FouT scan, restructured as chunked linear recurrence:
//   precompute (WMMA f32 GEMMs): A^2..A^512 (repeated squaring),
//     W[p] = (A^T)^{p+1} C  (doubling),  V[:,q] = A^q B (doubling),
//     k[m] = C . V[:,m]
//   per chunk g (64 chunks of 512 steps, stream-serialized):
//     pred[gK+p] = W[p].cprev + sum_{q<=p} k[p-q] u[gK+q] + D u[gK+p]
//     cnext      = A^512 cprev + sum_q V[:,511-q] u[gK+q]

#define NSTATE 1023
#define NPAD   1024
#define CHUNK  512
#define LSEQ   32768
#define NCHUNK 64

typedef __attribute__((ext_vector_type(2))) float v2f;
typedef __attribute__((ext_vector_type(8))) float v8f;

__device__ __forceinline__ v8f wmma4(v2f a, v2f b, v8f c) {
    return __builtin_amdgcn_wmma_f32_16x16x4_f32(
        /*neg_a=*/false, a, /*neg_b=*/false, b,
        /*c_mod=*/(short)0, c, /*reuse_a=*/false, /*reuse_b=*/false);
}

// ---------------- generic f32 WMMA GEMM (1 tile / wave), for tiny doubling levels ---------
__global__ void wmma_gemm_f32(const float* __restrict__ A, int lda,
                              const float* __restrict__ B, int ldb,
                              float* __restrict__ C, int ldc,
                              int Ncols, int Kdim, int totalTiles) {
    const int lane = threadIdx.x & 31;
    const int h    = lane >> 4;   // lane half
    const int r    = lane & 15;   // row/col within half
    const int ntiles = Ncols >> 4;
    const int tile = blockIdx.x * (blockDim.x >> 5) + (threadIdx.x >> 5);
    if (tile >= totalTiles) return;           // wave-uniform
    const int tm = tile / ntiles;
    const int tn = tile % ntiles;
    const int arow = tm * 16 + r;
    const int col  = tn * 16 + r;

    // pointer-bumped fragment addresses (no per-iter index math)
    const float* ap  = A + (size_t)arow * lda + 2 * h;            // A[arow][2h + kk]
    const float* bp0 = B + (size_t)(2 * h) * ldb + col;           // B[2h + kk][col]
    const float* bp1 = bp0 + ldb;                                 // B[2h+1+kk][col]
    const size_t bstep = 4 * (size_t)ldb;

    v8f acc = {};
    for (int kk = 0; kk < Kdim; kk += 4) {
        v2f a = *(const v2f*)ap;   // 8B-aligned: lda even, 2h even
        v2f b;
        b.x = bp0[0];
        b.y = bp1[0];
        acc = wmma4(a, b, acc);
        ap += 4; bp0 += bstep; bp1 += bstep;
    }
    float* Cp = C + (size_t)(tm * 16 + 8 * h) * ldc + col;   // D row = e + 8h
#pragma unroll
    for (int e = 0; e < 8; ++e) Cp[(size_t)e * ldc] = acc[e];
}

// ---------------- blocked f32 WMMA GEMM: 2x2 tiles (32x32 output) per wave ----------------
// A/B fragments reused across the 2x2 block: 6 VMEM per 4 WMMA instead of 3 per 1.
__global__ void wmma_gemm_f32_2x2(const float* __restrict__ A, int lda,
                                  const float* __restrict__ B, int ldb,
                                  float* __restrict__ C, int ldc,
                                  int Ncols, int Kdim, int totalTiles) {
    const int lane = threadIdx.x & 31;
    const int h    = lane >> 4;
    const int r    = lane & 15;
    const int ntiles = Ncols >> 5;            // macro-tiles (32 wide) along N
    const int tile = blockIdx.x * (blockDim.x >> 5) + (threadIdx.x >> 5);
    if (tile >= totalTiles) return;           // wave-uniform
    const int tm = tile / ntiles;
    const int tn = tile % ntiles;
    const int row0 = tm * 32 + r;
    const int col0 = tn * 32 + r;

    // pointer-bumped fragment addresses
    const float* ap0 = A + (size_t)row0 * lda + 2 * h;            // rows row0, row0+16
    const float* ap1 = ap0 + (size_t)16 * lda;
    const float* bp0 = B + (size_t)(2 * h) * ldb + col0;          // rows 2h+kk, 2h+1+kk
    const float* bp1 = bp0 + ldb;
    const size_t bstep = 4 * (size_t)ldb;

    v8f acc00 = {}, acc01 = {}, acc10 = {}, acc11 = {};
    for (int kk = 0; kk < Kdim; kk += 4) {
        v2f a0 = *(const v2f*)ap0;             // 8B-aligned
        v2f a1 = *(const v2f*)ap1;
        v2f b0, b1;
        b0.x = bp0[0];   b0.y = bp1[0];
        b1.x = bp0[16];  b1.y = bp1[16];
        acc00 = wmma4(a0, b0, acc00);
        acc01 = wmma4(a0, b1, acc01);
        acc10 = wmma4(a1, b0, acc10);
        acc11 = wmma4(a1, b1, acc11);
        ap0 += 4; ap1 += 4; bp0 += bstep; bp1 += bstep;
    }
    float* Cp0 = C + (size_t)(tm * 32 + 8 * h) * ldc + col0;        // rows e+8h
    float* Cp1 = C + (size_t)(tm * 32 + 16 + 8 * h) * ldc + col0;   // rows 16+e+8h
#pragma unroll
    for (int e = 0; e < 8; ++e) {
        Cp0[(size_t)e * ldc]      = acc00[e];
        Cp0[(size_t)e * ldc + 16] = acc01[e];
        Cp1[(size_t)e * ldc]      = acc10[e];
        Cp1[(size_t)e * ldc + 16] = acc11[e];
    }
}

// ---------------- init kernels ----------------
__global__ void pad_A_kernel(const float* __restrict__ A, float* __restrict__ Apow0) {
    int idx = blockIdx.x * blockDim.x + threadIdx.x;   // 1024*1024 threads
    int i = idx >> 10, j = idx & 1023;
    Apow0[idx] = (i < NSTATE && j < NSTATE) ? A[i * NSTATE + j] : 0.0f;
}

__global__ void pad_vec_kernel(const float* __restrict__ B, const float* __restrict__ C,
                               float* __restrict__ Bpad, float* __restrict__ Cpad,
                               float* __restrict__ c0, float* __restrict__ V) {
    int i = blockIdx.x * blockDim.x + threadIdx.x;     // 1024 threads
    float bv = (i < NSTATE) ? B[i] : 0.0f;
    Bpad[i] = bv;
    Cpad[i] = (i < NSTATE) ? C[i] : 0.0f;
    c0[i] = 0.0f;
    V[(size_t)i * CHUNK] = bv;                         // V[:,0] = A^0 B = B
}

// W[0][n] = (A^T C)[n]
__global__ void w1_kernel(const float* __restrict__ Apow0, const float* __restrict__ Cpad,
                          float* __restrict__ W) {
    int n = blockIdx.x * blockDim.x + threadIdx.x;     // 1024 threads
    float s = 0.0f;
    for (int m = 0; m < NPAD; ++m) s += Apow0[(size_t)m * NPAD + n] * Cpad[m];
    W[n] = s;
}

// k[m] = C . V[:,m]
__global__ void kvec_kernel(const float* __restrict__ Cpad, const float* __restrict__ V,
                            float* __restrict__ kvec) {
    int m = blockIdx.x * blockDim.x + threadIdx.x;     // 512 threads
    float s = 0.0f;
    for (int n = 0; n < NPAD; ++n) s += Cpad[n] * V[(size_t)n * CHUNK + m];
    kvec[m] = s;
}

// ---------------- per-chunk kernels (one wave per output element) ----------------
__global__ void pred_kernel(const float* __restrict__ W, const float* __restrict__ kvec,
                            const float* __restrict__ cprev, const float* __restrict__ u,
                            const float* __restrict__ Dd, float* __restrict__ pred, int base) {
    int wid  = (blockIdx.x * blockDim.x + threadIdx.x) >> 5;  // p = 0..511
    int lane = threadIdx.x & 31;
    if (wid >= CHUNK) return;
    const float* wrow = W + (size_t)wid * NPAD;
    float s = 0.0f;
    for (int n = lane; n < NPAD; n += 32) s += wrow[n] * cprev[n];
    for (int q = lane; q <= wid; q += 32) s += kvec[wid - q] * u[base + q];
    for (int off = 16; off; off >>= 1) s += __shfl_xor(s, off, 32);
    if (lane == 0) pred[base + wid] = s + Dd[0] * u[base + wid];
}

__global__ void cnew_kernel(const float* __restrict__ A512, const float* __restrict__ V,
                            const float* __restrict__ cprev, const float* __restrict__ u,
                            float* __restrict__ cnext, int base) {
    int wid  = (blockIdx.x * blockDim.x + threadIdx.x) >> 5;  // n = 0..1023
    int lane = threadIdx.x & 31;
    if (wid >= NPAD) return;
    const float* arow = A512 + (size_t)wid * NPAD;
    float s = 0.0f;
    for (int n = lane; n < NPAD; n += 32) s += arow[n] * cprev[n];
    const float* vrow = V + (size_t)wid * CHUNK;
    for (int q = lane; q < CHUNK; q += 32) s += vrow[CHUNK - 1 - q] * u[base + q];
    for (int off = 16; off; off >>= 1) s += __shfl_xor(s, off, 32);
    if (lane == 0) cnext[wid] = s;
}

extern "C" void kernel_launch(void* const* d_in, const int* in_sizes, int n_in,
                              void* d_out, int out_size, void* d_ws, size_t ws_size,
                              hipStream_t stream) {
    const float* u = (const float*)d_in[0];   // (L,)
    const float* A = (const float*)d_in[1];   // (N,N)
    const float* B = (const float*)d_in[2];   // (N,)
    const float* C = (const float*)d_in[3];   // (N,)
    const float* D = (const float*)d_in[4];   // (1,)
    float* pred = (float*)d_out;              // (L,1) flat

    // workspace layout (floats): ~12.6 MB total
    float* ws = (float*)d_ws;
    float* Apow[2] = { ws, ws + (size_t)NPAD * NPAD };
    float* W    = ws + 2 * (size_t)NPAD * NPAD;        // [512][1024], row p = w_{p+1}
    float* V    = W + (size_t)CHUNK * NPAD;            // [1024][512], col q = A^q B
    float* kvec = V + (size_t)NPAD * CHUNK;            // [512]
    float* Bpad = kvec + CHUNK;                        // [1024]
    float* Cpad = Bpad + NPAD;                         // [1024]
    float* cb0  = Cpad + NPAD;                         // [1024]
    float* cb1  = cb0 + NPAD;                          // [1024]

    pad_A_kernel<<<(NPAD * NPAD) / 256, 256, 0, stream>>>(A, Apow[0]);
    pad_vec_kernel<<<NPAD / 256, 256, 0, stream>>>(B, C, Bpad, Cpad, cb0, V);
    w1_kernel<<<NPAD / 256, 256, 0, stream>>>(Apow[0], Cpad, W);

    // GEMM dispatch: blocked 2x2 kernel when dims allow (all big GEMMs), 1-tile otherwise.
    auto launch_gemm = [&](const float* Am, int lda, const float* Bm, int ldb,
                           float* Cm, int ldc, int M, int Ncols, int Kdim) {
        if ((M & 31) == 0 && (Ncols & 31) == 0) {
            int tiles = (M / 32) * (Ncols / 32);
            wmma_gemm_f32_2x2<<<(tiles + 7) / 8, 256, 0, stream>>>(
                Am, lda, Bm, ldb, Cm, ldc, Ncols, Kdim, tiles);
        } else {
            int tiles = (M / 16) * (Ncols / 16);
            wmma_gemm_f32<<<(tiles + 7) / 8, 256, 0, stream>>>(
                Am, lda, Bm, ldb, Cm, ldc, Ncols, Kdim, tiles);
        }
    };

    // doubling: at level m we hold A^m; extend V cols [m,2m), W rows [m,2m); square.
    int cur = 0;
    for (int m = 1; m <= 256; m <<= 1) {
        // V[:, m:2m) = A^m * V[:, 0:m)   (tile-padded to 16 cols at small m)
        int vN = (m < 16) ? 16 : m;
        launch_gemm(Apow[cur], NPAD, V, CHUNK, V + m, CHUNK, NPAD, vN, NPAD);
        // W[m:2m, :] = W[0:m, :] * A^m   (w_{m+i} = (A^m)^T w_i, row-vector form)
        int wM = (m < 16) ? 16 : m;
        launch_gemm(W, NPAD, Apow[cur], NPAD, W + (size_t)m * NPAD, NPAD, wM, NPAD, NPAD);
        // A^{2m} = A^m * A^m
        launch_gemm(Apow[cur], NPAD, Apow[cur], NPAD, Apow[1 - cur], NPAD, NPAD, NPAD, NPAD);
        cur ^= 1;
    }
    const float* A512 = Apow[cur];

    kvec_kernel<<<CHUNK / 256, 256, 0, stream>>>(Cpad, V, kvec);

    // 64 stream-serialized chunks; ping-pong state buffers
    for (int g = 0; g < NCHUNK; ++g) {
        float* cp = (g & 1) ? cb1 : cb0;
        float* cn = (g & 1) ? cb0 : cb1;
        int base = g * CHUNK;
        pred_kernel<<<(CHUNK * 32) / 256, 256, 0, stream>>>(W, kvec, cp, u, D, pred, base);
        cnew_kernel<<<(NPAD * 32) / 256, 256, 0, stream>>>(A512, V, cp, u, cn, base);
    }
}